// MonitorRetina_67508295958939
// MI455X (gfx1250) — compile-verified
//
#include <hip/hip_runtime.h>
#include <hip/hip_bf16.h>
#include <math.h>

typedef float v2f __attribute__((ext_vector_type(2)));
typedef float v8f __attribute__((ext_vector_type(8)));

#define NB     256
#define HH     144
#define WW     256
#define HW     (HH * WW)          // 36864
#define FDIM   16
#define DDIM   128

__device__ __constant__ const float kFovHalf   = 0.65449846949787352f;  // 0.5*75*pi/180
__device__ __constant__ const float kGeluGamma = 1.7015043497085571f;

__device__ __forceinline__ float gelu_tanh_scaled(float x) {
    const float c = 0.7978845608028654f; // sqrt(2/pi)
    float x3 = x * x * x;
    float t  = tanhf(c * (x + 0.044715f * x3));
    return 0.5f * x * (1.0f + t) * kGeluGamma;
}

// ---------------------------------------------------------------------------
// GEMM1: h1 = gelu(P[256x16] @ W1[16x128] + b1) * gamma, via V_WMMA_F32_16X16X4_F32
// One wave per 16x16 output tile. grid = (8 ntiles, 16 mtiles), block = 32.
// ---------------------------------------------------------------------------
__global__ __launch_bounds__(32) void gemm1_wmma(const float* __restrict__ P,
                                                 const float* __restrict__ W1,
                                                 const float* __restrict__ b1,
                                                 float* __restrict__ h1) {
    const int nt = blockIdx.x, mt = blockIdx.y;
    const int lane = threadIdx.x & 31;
    const int g = lane >> 4;      // half-wave group
    const int l = lane & 15;      // row (A) / col (B,C) within tile
    const int m0 = mt * 16, n0 = nt * 16;

    v8f acc = {};
#pragma unroll
    for (int k0 = 0; k0 < FDIM; k0 += 4) {
        const int ka = k0 + 2 * g;
        v2f A, B;
        A.x = P[(m0 + l) * FDIM + ka];
        A.y = P[(m0 + l) * FDIM + ka + 1];
        B.x = W1[ka * DDIM + n0 + l];
        B.y = W1[(ka + 1) * DDIM + n0 + l];
        acc = __builtin_amdgcn_wmma_f32_16x16x4_f32(false, A, false, B,
                                                    (short)0, acc, false, false);
    }
    const float bias = b1[n0 + l];
#pragma unroll
    for (int r = 0; r < 8; ++r) {
        const int m = m0 + r + 8 * g;   // C layout: vgpr r holds M=r (lanes 0-15), M=r+8 (lanes 16-31)
        h1[m * DDIM + n0 + l] = gelu_tanh_scaled(acc[r] + bias);
    }
}

// ---------------------------------------------------------------------------
// GEMM2: h2 = gelu(h1[256x128] @ W2[128x128] + b2) * gamma
// ---------------------------------------------------------------------------
__global__ __launch_bounds__(32) void gemm2_wmma(const float* __restrict__ h1,
                                                 const float* __restrict__ W2,
                                                 const float* __restrict__ b2,
                                                 float* __restrict__ h2) {
    const int nt = blockIdx.x, mt = blockIdx.y;
    const int lane = threadIdx.x & 31;
    const int g = lane >> 4;
    const int l = lane & 15;
    const int m0 = mt * 16, n0 = nt * 16;

    v8f acc = {};
    for (int k0 = 0; k0 < DDIM; k0 += 4) {
        const int ka = k0 + 2 * g;
        v2f A, B;
        A.x = h1[(m0 + l) * DDIM + ka];
        A.y = h1[(m0 + l) * DDIM + ka + 1];
        B.x = W2[ka * DDIM + n0 + l];
        B.y = W2[(ka + 1) * DDIM + n0 + l];
        acc = __builtin_amdgcn_wmma_f32_16x16x4_f32(false, A, false, B,
                                                    (short)0, acc, false, false);
    }
    const float bias = b2[n0 + l];
#pragma unroll
    for (int r = 0; r < 8; ++r) {
        const int m = m0 + r + 8 * g;
        h2[m * DDIM + n0 + l] = gelu_tanh_scaled(acc[r] + bias);
    }
}

// ---------------------------------------------------------------------------
// Head: ang = h2 @ Wp[128x3] + bp; build R = Rz@Ry@Rx per batch. 256 threads.
// ---------------------------------------------------------------------------
__global__ void head_rot(const float* __restrict__ h2,
                         const float* __restrict__ Wp,
                         const float* __restrict__ bp,
                         float* __restrict__ Rout) {
    const int t = blockIdx.x * blockDim.x + threadIdx.x;
    if (t >= NB) return;
    float a0 = bp[0], a1 = bp[1], a2 = bp[2];
    for (int k = 0; k < DDIM; ++k) {
        const float h = h2[t * DDIM + k];
        a0 = fmaf(h, Wp[k * 3 + 0], a0);
        a1 = fmaf(h, Wp[k * 3 + 1], a1);
        a2 = fmaf(h, Wp[k * 3 + 2], a2);
    }
    const float cx = cosf(a0), sx = sinf(a0);
    const float cy = cosf(a1), sy = sinf(a1);
    const float cz = cosf(a2), sz = sinf(a2);
    float* R = Rout + t * 9;
    R[0] = cz * cy;  R[1] = -sz * cx + cz * sy * sx;  R[2] = sz * sx + cz * sy * cx;
    R[3] = sz * cy;  R[4] =  cz * cx + sz * sy * sx;  R[5] = -cz * sx + sz * sy * cx;
    R[6] = -sy;      R[7] =  cy * sx;                 R[8] = cy * cx;
}

// ---------------------------------------------------------------------------
// Precompute fixed retina directions for the 144x256 grid (batch-invariant).
// ---------------------------------------------------------------------------
__global__ void dirs_kernel(float* __restrict__ dx, float* __restrict__ dy,
                            float* __restrict__ dz) {
    const int p = blockIdx.x * blockDim.x + threadIdx.x;
    if (p >= HW) return;
    const int y = p >> 8, x = p & 255;
    const float gx = (float)(2 * x + 1 - WW) * (1.0f / 256.0f);
    const float gy = (float)(2 * y + 1 - HH) * (1.0f / 256.0f);
    const float ax = gx * kFovHalf, ay = gy * kFovHalf;
    const float a = sqrtf(ax * ax + ay * ay) + 1e-12f;
    const float s = sinf(a) / a;
    dx[p] = s * ax;
    dy[p] = s * ay;
    dz[p] = cosf(a);
}

// ---------------------------------------------------------------------------
// Render: rotate dirs by R[n], project, bilinear sample stimulus[n].
// 4 pixels per thread -> b128 dir loads and b128 output stores.
// grid = (HW/4/256, NB), block = 256.
// ---------------------------------------------------------------------------
__device__ __forceinline__ float tap(const float* __restrict__ img, int y, int x) {
    const bool valid = (x >= 0) & (x < WW) & (y >= 0) & (y < HH);
    const int yc = min(max(y, 0), HH - 1);
    const int xc = min(max(x, 0), WW - 1);
    const float v = img[yc * WW + xc];
    return valid ? v : 0.0f;
}

__global__ __launch_bounds__(256) void render_kernel(const float* __restrict__ stim,
                                                     const float* __restrict__ Rmat,
                                                     const float* __restrict__ dirx,
                                                     const float* __restrict__ diry,
                                                     const float* __restrict__ dirz,
                                                     float* __restrict__ out) {
    const int n = blockIdx.y;
    const int quad = blockIdx.x * blockDim.x + threadIdx.x;
    const int p = quad * 4;
    if (p >= HW) return;

    const float* __restrict__ R = Rmat + n * 9;       // uniform per block -> SGPRs
    const float r00 = R[0], r01 = R[1], r02 = R[2];
    const float r10 = R[3], r11 = R[4], r12 = R[5];
    const float r20 = R[6], r21 = R[7], r22 = R[8];

    const float4 dx4 = *(const float4*)(dirx + p);
    const float4 dy4 = *(const float4*)(diry + p);
    const float4 dz4 = *(const float4*)(dirz + p);
    const float* __restrict__ img = stim + (size_t)n * HW;

    float res[4];
#pragma unroll
    for (int i = 0; i < 4; ++i) {
        const float dx = (&dx4.x)[i], dy = (&dy4.x)[i], dz = (&dz4.x)[i];
        const float rx = r00 * dx + r01 * dy + r02 * dz;
        const float ry = r10 * dx + r11 * dy + r12 * dz;
        const float rz = r20 * dx + r21 * dy + r22 * dz;
        float gx, gy;
        if (rz > 0.001f) {
            const float inv = 1.0f / rz;
            gx = rx * inv;
            gy = ry * inv;
        } else {
            gx = 10.0f;
            gy = 10.0f;
        }
        const float px = (gx * 256.0f + (float)(WW - 1)) * 0.5f;
        const float py = (gy * 256.0f + (float)(HH - 1)) * 0.5f;
        const float x0f = floorf(px), y0f = floorf(py);
        const float wx = px - x0f, wy = py - y0f;
        const int x0 = (int)x0f, y0 = (int)y0f;
        const float v00 = tap(img, y0, x0);
        const float v01 = tap(img, y0, x0 + 1);
        const float v10 = tap(img, y0 + 1, x0);
        const float v11 = tap(img, y0 + 1, x0 + 1);
        res[i] = (v00 * (1.0f - wx) + v01 * wx) * (1.0f - wy) +
                 (v10 * (1.0f - wx) + v11 * wx) * wy;
    }
    *(float4*)(out + (size_t)n * HW + p) = make_float4(res[0], res[1], res[2], res[3]);
}

// ---------------------------------------------------------------------------
extern "C" void kernel_launch(void* const* d_in, const int* in_sizes, int n_in,
                              void* d_out, int out_size, void* d_ws, size_t ws_size,
                              hipStream_t stream) {
    (void)in_sizes; (void)n_in; (void)out_size; (void)ws_size;
    const float* stimulus    = (const float*)d_in[0];
    const float* perspective = (const float*)d_in[1];
    const float* W1          = (const float*)d_in[2];
    const float* b1          = (const float*)d_in[3];
    const float* W2          = (const float*)d_in[4];
    const float* b2          = (const float*)d_in[5];
    const float* Wp          = (const float*)d_in[6];
    const float* bp          = (const float*)d_in[7];
    float* out = (float*)d_out;

    float* ws   = (float*)d_ws;
    float* h1   = ws;                       // 256*128
    float* h2   = ws + 32768;               // 256*128
    float* R    = ws + 65536;               // 256*9
    float* dirx = ws + 67840;               // 36864
    float* diry = dirx + HW;
    float* dirz = diry + HW;

    gemm1_wmma<<<dim3(DDIM / 16, NB / 16), 32, 0, stream>>>(perspective, W1, b1, h1);
    gemm2_wmma<<<dim3(DDIM / 16, NB / 16), 32, 0, stream>>>(h1, W2, b2, h2);
    head_rot<<<1, 256, 0, stream>>>(h2, Wp, bp, R);
    dirs_kernel<<<HW / 256, 256, 0, stream>>>(dirx, diry, dirz);
    render_kernel<<<dim3(HW / 4 / 256, NB), 256, 0, stream>>>(stimulus, R, dirx, diry,
                                                              dirz, out);
}